// TopPool_68710886801776
// MI455X (gfx1250) — compile-verified
//
#include <hip/hip_runtime.h>

// TopPool: out[b,c,h,w] = max_{h' >= h} x[b,c,h',w]  (reverse cummax over H)
// x: (32, 256, 128, 128) fp32  ->  8192 independent (H=128, W=128) planes.
// Pure streaming op: 1 GiB compulsory traffic, ~46 us floor at 23.3 TB/s.

typedef float v4f __attribute__((ext_vector_type(4)));

#define H_DIM   128
#define W_DIM   128
#define ROW_V4  (W_DIM / 4)   // 32 float4 per row
#define UNROLL  8             // 8 x B128 loads in flight per wave per chunk

__device__ __forceinline__ v4f vmax4(v4f a, v4f b) {
    v4f r;
    r.x = __builtin_fmaxf(a.x, b.x);
    r.y = __builtin_fmaxf(a.y, b.y);
    r.z = __builtin_fmaxf(a.z, b.z);
    r.w = __builtin_fmaxf(a.w, b.w);
    return r;
}

__global__ __launch_bounds__(256) void toppool_suffix_max(
        const float* __restrict__ x, float* __restrict__ y, int planes) {
    // One thread owns one 4-wide W column of one (b,c) plane and walks H
    // top-down keeping a running max. 32 consecutive lanes cover one full
    // 512-byte row of a plane -> perfectly coalesced wave32 B128 traffic.
    const int t  = blockIdx.x * blockDim.x + threadIdx.x;
    const int p  = t >> 5;          // plane index (b*C + c)
    if (p >= planes) return;
    const int w4 = t & 31;          // float4 column within the row

    const size_t base = (size_t)p * (H_DIM * W_DIM) + (size_t)w4 * 4;
    const v4f* __restrict__ xin  = (const v4f*)(x + base);
    v4f*       __restrict__ yout = (v4f*)(y + base);

    const float ninf = -__builtin_inff();
    v4f run = (v4f){ninf, ninf, ninf, ninf};

    // h = 127 .. 0 in chunks of UNROLL. Loads are independent and issued as a
    // batch (B128, non-temporal: each element is touched exactly once); the
    // serial max chain then folds them in order, storing each running max.
    for (int hc = H_DIM; hc > 0; hc -= UNROLL) {
        // Prefetch the first row of the chunk two ahead of the descending
        // stream (gfx1250 global_prefetch_b8); uniform branch, no divergence.
        if (hc > 2 * UNROLL) {
            __builtin_prefetch(&xin[(size_t)(hc - 1 - 2 * UNROLL) * ROW_V4], 0, 3);
        }

        v4f v[UNROLL];
        #pragma unroll
        for (int u = 0; u < UNROLL; ++u) {
            v[u] = __builtin_nontemporal_load(&xin[(size_t)(hc - 1 - u) * ROW_V4]);
        }

        #pragma unroll
        for (int u = 0; u < UNROLL; ++u) {
            run = vmax4(run, v[u]);
            __builtin_nontemporal_store(run, &yout[(size_t)(hc - 1 - u) * ROW_V4]);
        }
    }
}

extern "C" void kernel_launch(void* const* d_in, const int* in_sizes, int n_in,
                              void* d_out, int out_size, void* d_ws, size_t ws_size,
                              hipStream_t stream) {
    const float* x = (const float*)d_in[0];
    float*       y = (float*)d_out;

    const int n      = in_sizes[0];              // 32*256*128*128 = 134217728
    const int planes = n / (H_DIM * W_DIM);      // 8192
    const int total  = planes * ROW_V4;          // 262144 threads (8192 waves)

    const int block = 256;                       // 8 wave32 per workgroup
    const int grid  = (total + block - 1) / block;

    toppool_suffix_max<<<grid, block, 0, stream>>>(x, y, planes);
}